// ToFU_38104949850619
// MI455X (gfx1250) — compile-verified
//
#include <hip/hip_runtime.h>
#include <hip/hip_bf16.h>

#define N    512
#define TILE 16
#define NT   512   // threads in the Hungarian block (16 wave32s)

typedef __attribute__((ext_vector_type(2))) float v2f;
typedef __attribute__((ext_vector_type(8))) float v8f;

// ---------------------------------------------------------------------------
// Kernel 1: 512x512 Euclidean cost matrix via V_WMMA_F32_16X16X4_F32.
// One wave (32 threads) per 16x16 tile. dist(i,j) = sqrt(|p|^2+|q|^2-2 p.q),
// with the Gram term p.q computed by the f32 WMMA (K=2 zero-padded to 4).
// A layout (16x4 f32, 2 VGPRs/lane): lanes 0-15 hold {K0,K1} of row M=lane,
// lanes 16-31 hold {K2,K3} (= zeros). B (4x16) mirrors that per ISA 7.12.2.
// D layout: VGPR r -> row M = r + 8*(lane>=16), col N = lane&15.
// ---------------------------------------------------------------------------
__global__ __launch_bounds__(32)
void tofu_cost_wmma(const float* __restrict__ dgm,
                    const float* __restrict__ dgmx,
                    float* __restrict__ cost) {
    const int lane = threadIdx.x;      // 0..31, full wave -> EXEC all ones
    const int tI   = blockIdx.x;       // row tile
    const int tJ   = blockIdx.y;       // col tile
    const int half = lane >> 4;        // 0: carries K=0,1 ; 1: K=2,3 (zeros)
    const int l15  = lane & 15;

    v2f a; v2f b;
    if (half == 0) {
        const int row = tI * TILE + l15;
        const int col = tJ * TILE + l15;
        a.x = dgm [2 * row + 0];
        a.y = dgm [2 * row + 1];
        b.x = dgmx[2 * col + 0];
        b.y = dgmx[2 * col + 1];
    } else {
        a.x = 0.0f; a.y = 0.0f; b.x = 0.0f; b.y = 0.0f;
    }

    v8f c = {};
    // 8 args: (neg_a, A, neg_b, B, c_mod, C, reuse_a, reuse_b)
    c = __builtin_amdgcn_wmma_f32_16x16x4_f32(false, a, false, b,
                                              (short)0, c, false, false);

    // This lane's output column data
    const int   col = tJ * TILE + l15;
    const float qb  = dgmx[2 * col + 0];
    const float qd  = dgmx[2 * col + 1];
    const float nq  = qb * qb + qd * qd;

#pragma unroll
    for (int r = 0; r < 8; ++r) {
        const int   row = tI * TILE + r + half * 8;
        const float pb  = dgm[2 * row + 0];
        const float pd  = dgm[2 * row + 1];
        const float d2  = fmaxf(pb * pb + pd * pd + nq - 2.0f * c[r], 0.0f);
        cost[row * N + col] = sqrtf(d2);
    }
}

// ---------------------------------------------------------------------------
// Kernel 2: Jonker-Volgenant assignment (mirrors the numpy reference) in one
// workgroup. All state in LDS; 512 threads relax one column each; argmin via
// LDS tree reduction with first-index tie-break (matches np.argmin). Then the
// final loss 0.5 * sum ||dgm[p[j]-1] - dgm_x[j-1]||^2 is reduced and stored.
// ---------------------------------------------------------------------------
__global__ __launch_bounds__(NT)
void tofu_hungarian(const float* __restrict__ cost,
                    const float* __restrict__ dgm,
                    const float* __restrict__ dgmx,
                    float* __restrict__ out) {
    __shared__ float u[N + 1], v[N + 1], minv[N + 1];
    __shared__ int   p[N + 1], way[N + 1];
    __shared__ int   used[N + 1];
    __shared__ float red_val[NT];
    __shared__ int   red_idx[NT];
    __shared__ int   s_j0, s_i0, s_done;

    const float INF = __builtin_inff();
    const int   tid = threadIdx.x;
    const int   j   = tid + 1;        // my column, 1..N (1-indexed like ref)

    u[tid] = 0.0f; v[tid] = 0.0f; p[j] = 0; way[j] = 0;
    if (tid == 0) { u[N] = 0.0f; v[N] = 0.0f; p[0] = 0; way[0] = 0; }
    __syncthreads();

    for (int i = 1; i <= N; ++i) {
        minv[j] = INF; used[j] = 0;
        if (tid == 0) { p[0] = i; s_j0 = 0; used[0] = 0; minv[0] = INF; }
        __syncthreads();

        while (true) {
            if (tid == 0) { used[s_j0] = 1; s_i0 = p[s_j0]; }
            __syncthreads();
            const int i0 = s_i0;
            const int j0 = s_j0;

            // Relax my column (only thread-local writes to minv/way)
            if (!used[j]) {
                const float cur = cost[(i0 - 1) * N + (j - 1)] - u[i0] - v[j];
                if (cur < minv[j]) { minv[j] = cur; way[j] = j0; }
            }
            red_val[tid] = used[j] ? INF : minv[j];
            red_idx[tid] = j;
            __syncthreads();

            // argmin over free columns; ties -> smallest j (np.argmin)
            for (int s = NT / 2; s > 0; s >>= 1) {
                if (tid < s) {
                    const float bv = red_val[tid + s];
                    if (bv < red_val[tid] ||
                        (bv == red_val[tid] && red_idx[tid + s] < red_idx[tid])) {
                        red_val[tid] = bv;
                        red_idx[tid] = red_idx[tid + s];
                    }
                }
                __syncthreads();
            }
            const float delta = red_val[0];
            const int   j1    = red_idx[0];

            // Dual updates: used cols -> u[p[jj]] += delta (distinct rows,
            // race-free), v[jj] -= delta; free cols -> minv[jj] -= delta.
            {
                int jj = tid;                      // covers 0..NT-1
                if (used[jj]) { u[p[jj]] += delta; v[jj] -= delta; }
                else          { minv[jj] -= delta; }
                if (tid == NT - 1) {               // also cover jj = N
                    jj = N;
                    if (used[jj]) { u[p[jj]] += delta; v[jj] -= delta; }
                    else          { minv[jj] -= delta; }
                }
            }
            if (tid == 0) { s_j0 = j1; s_done = (p[j1] == 0); }
            __syncthreads();
            if (s_done) break;
        }

        // Augment along the alternating tree (sequential, one thread)
        if (tid == 0) {
            int j0 = s_j0;
            while (j0) {
                const int j1 = way[j0];
                p[j0] = p[j1];
                j0 = j1;
            }
        }
        __syncthreads();
    }

    // loss = 0.5 * sum over columns j of ||dgm[p[j]-1] - dgm_x[j-1]||^2
    {
        const int   row  = p[j] - 1;
        const int   colx = j - 1;
        const float db   = dgm[2 * row + 0] - dgmx[2 * colx + 0];
        const float dd   = dgm[2 * row + 1] - dgmx[2 * colx + 1];
        red_val[tid] = db * db + dd * dd;
        __syncthreads();
        for (int s = NT / 2; s > 0; s >>= 1) {
            if (tid < s) red_val[tid] += red_val[tid + s];
            __syncthreads();
        }
        if (tid == 0) out[0] = 0.5f * red_val[0];
    }
}

extern "C" void kernel_launch(void* const* d_in, const int* in_sizes, int n_in,
                              void* d_out, int out_size, void* d_ws, size_t ws_size,
                              hipStream_t stream) {
    const float* dgm  = (const float*)d_in[0];   // [512, 2] float32
    const float* dgmx = (const float*)d_in[1];   // [512, 2] float32
    float*       cost = (float*)d_ws;            // 512*512*4 = 1 MB scratch

    dim3 grid(N / TILE, N / TILE);               // 32x32 tiles, 1 wave each
    tofu_cost_wmma<<<grid, dim3(32), 0, stream>>>(dgm, dgmx, cost);
    tofu_hungarian<<<dim3(1), dim3(NT), 0, stream>>>(cost, dgm, dgmx, (float*)d_out);
}